// LocalAttention_17463337026222
// MI455X (gfx1250) — compile-verified
//
#include <hip/hip_runtime.h>
#include <hip/hip_bf16.h>

// Problem constants (from reference)
#define BB   16
#define SS   2048
#define DD   512
#define WW   5
#define KK   11           // 2*w+1
#define TS   16           // token positions per block tile
#define HALO (TS + 2*WW)  // 26 rows staged per tile
#define XSTR 520          // padded LDS row stride (floats) -> conflict-free across lane halves
#define ASTR 33           // padded Aexp row stride (floats)
#define AK   28           // K dim padded to multiple of 4 (>= HALO=26)

typedef float v2f __attribute__((ext_vector_type(2)));
typedef float v8f __attribute__((ext_vector_type(8)));

__global__ __launch_bounds__(256) void local_attn_wmma_kernel(
    const float* __restrict__ X,     // [B,S,D]
    const float* __restrict__ Wv,    // [D]
    const float* __restrict__ bv,    // [1]
    float* __restrict__ Out)         // [B,S,D]
{
    __shared__ float xs[HALO * XSTR];   // staged halo tile  (~53 KB)
    __shared__ float sbuf[HALO];        // scores for halo rows
    __shared__ float Aexp[TS * ASTR];   // expanded banded alpha (16 x AK used)

    const int tid  = threadIdx.x;
    const int t0   = blockIdx.x * TS;
    const int b    = blockIdx.y;
    const float* Xb = X + (size_t)b * SS * DD;

    // ---------------- Phase 0: stage halo rows [t0-5, t0+21) into LDS ----------------
    // 26 rows * 128 float4 = 3328 vec4 loads, 13 per thread. Clamped rows are
    // multiplied by alpha==0 later, so clamping is safe.
    for (int v = tid; v < HALO * (DD / 4); v += 256) {
        int r  = v / (DD / 4);
        int c4 = (v % (DD / 4)) * 4;
        int gr = t0 - WW + r;
        gr = gr < 0 ? 0 : (gr >= SS ? SS - 1 : gr);
        float4 val = *(const float4*)(Xb + (size_t)gr * DD + c4);
        *(float4*)(&xs[r * XSTR + c4]) = val;
    }
    for (int v = tid; v < TS * ASTR; v += 256) Aexp[v] = 0.0f;
    __syncthreads();

    // ---------------- Phase 1: scores s[r] = dot(xs[r,:], W) + b ----------------
    const int wave = tid >> 5;
    const int lane = tid & 31;
    for (int r = wave; r < HALO; r += 8) {
        float p = 0.0f;
        #pragma unroll
        for (int i = 0; i < DD / 32; ++i)
            p = fmaf(xs[r * XSTR + lane + 32 * i], Wv[lane + 32 * i], p);
        #pragma unroll
        for (int off = 16; off > 0; off >>= 1)
            p += __shfl_xor(p, off, 32);
        if (lane == 0) sbuf[r] = p + bv[0];
    }
    __syncthreads();

    // ---------------- Phase 2: masked band softmax -> Aexp[i][i+k] ----------------
    if (tid < TS) {
        const int i = tid;
        float band[KK];
        float m = -__builtin_inff();
        #pragma unroll
        for (int k = 0; k < KK; ++k) {
            int raw  = t0 + i - WW + k;
            bool ok  = (raw >= 0) && (raw < SS);
            float v  = ok ? sbuf[i + k] : -__builtin_inff();
            band[k]  = v;
            m = v > m ? v : m;
        }
        float sum = 0.0f;
        #pragma unroll
        for (int k = 0; k < KK; ++k) {
            float e = __expf(band[k] - m);   // exp(-inf) == 0 for masked entries
            band[k] = e;
            sum += e;
        }
        float inv = 1.0f / sum;
        #pragma unroll
        for (int k = 0; k < KK; ++k)
            Aexp[i * ASTR + (i + k)] = band[k] * inv;
    }
    __syncthreads();

    // ---------------- Phase 3: out_tile = Aexp(16xAK) x xs(AKx16) via WMMA f32 ----------------
    // f32 16x16x4 operand layout (wave32):
    //   A: lanes 0-15 -> M=0..15, K = 4s + 2*(lane>=16) + vgpr
    //   B: lane halves split K the same way, N = lane&15 striped in each VGPR
    //   C/D: vgpr v -> row v + 8*(lane>=16), col = lane&15
    const int Llo = lane & 15;
    const int Lhi = lane >> 4;
    float* Ob = Out + ((size_t)b * SS + t0) * DD;

    for (int chunk = wave; chunk < DD / 16; chunk += 8) {
        const int d0 = chunk * 16;
        v8f acc = {};
        #pragma unroll
        for (int s = 0; s < AK / 4; ++s) {
            const int k = s * 4 + 2 * Lhi;
            v2f a, bm;
            a.x = Aexp[Llo * ASTR + k];
            a.y = Aexp[Llo * ASTR + k + 1];
            // branchless zero-mask for padded K rows (26,27); Aexp is 0 there too
            const int r0 = (k     < HALO) ? k     : 0;
            const int r1 = (k + 1 < HALO) ? k + 1 : 0;
            const float m0 = (k     < HALO) ? 1.0f : 0.0f;
            const float m1 = (k + 1 < HALO) ? 1.0f : 0.0f;
            bm.x = m0 * xs[r0 * XSTR + d0 + Llo];
            bm.y = m1 * xs[r1 * XSTR + d0 + Llo];
            acc = __builtin_amdgcn_wmma_f32_16x16x4_f32(
                false, a, false, bm, (short)0, acc, false, false);
        }
        #pragma unroll
        for (int v = 0; v < 8; ++v)
            Ob[(size_t)(v + 8 * Lhi) * DD + d0 + Llo] = acc[v];
    }
}

extern "C" void kernel_launch(void* const* d_in, const int* in_sizes, int n_in,
                              void* d_out, int out_size, void* d_ws, size_t ws_size,
                              hipStream_t stream) {
    const float* X  = (const float*)d_in[0];   // [B,S,D] fp32
    const float* Wv = (const float*)d_in[1];   // [D]
    const float* bv = (const float*)d_in[2];   // [1]
    float* Out = (float*)d_out;                // [B,S,D] fp32

    dim3 grid(SS / TS, BB);   // 128 x 16 blocks
    dim3 block(256);          // 8 waves (wave32)
    local_attn_wmma_kernel<<<grid, block, 0, stream>>>(X, Wv, bv, Out);
}